// TritonDynamicLinear_60327110639757
// MI455X (gfx1250) — compile-verified
//
#include <hip/hip_runtime.h>
#include <hip/hip_bf16.h>

// ---------------------------------------------------------------------------
// Block-sparse DDS GEMM for MI455X (gfx1250, wave32, WMMA).
//   out[M=4096, N=4096] = data[M, K=4096] @ W^T, W[N, K] masked at 32x32
//   block granularity (~50% blocks active).
//
// Pipeline (CDNA5-idiomatic):
//   * 128x128 block mask precomputed; per-k-block skip is workgroup-uniform.
//   * Staging uses GLOBAL_LOAD_ASYNC_TO_LDS_B128 (ASYNCcnt) into ping-pong
//     LDS buffers: iteration i computes on buf while the async engine fills
//     buf^1 for the NEXT ACTIVE k-block. One barrier per iteration.
//   * Each of 8 waves computes a 32x32 tile via 2x2 V_WMMA_F32_16X16X4_F32
//     accumulators; LDS stride 36 floats -> 16B-aligned async writes AND
//     conflict-free b64 fragment reads (36*r mod 64 covers all mult-of-4
//     residues for r=0..15).
// ---------------------------------------------------------------------------

typedef __attribute__((ext_vector_type(2))) float v2f;
typedef __attribute__((ext_vector_type(8))) float v8f;

#define M_DIM 4096
#define N_DIM 4096
#define K_DIM 4096
#define BH 32
#define BW 32
#define NB (N_DIM / BH)   // 128 n-blocks
#define KB (K_DIM / BW)   // 128 k-blocks

#define TILE_M 256        // workgroup M tile (8 waves x 32 rows)
#define TILE_N 32         // workgroup N tile (one n-block)
#define TILE_K 32         // one k-block per staged chunk
#define LDS_STRIDE 36     // floats; 144B row pitch: 16B-aligned, bank-conflict-free

// Async copy of 16 bytes from global memory directly into LDS (no VGPR
// round-trip). Tracked by ASYNCcnt. Low 32 bits of a generic pointer to a
// __shared__ object are the wave-relative LDS byte offset.
__device__ __forceinline__ void async_copy_b128(const float* lds_dst,
                                                const float* gsrc) {
  unsigned lds_off = (unsigned)(unsigned long long)(const void*)lds_dst;
  asm volatile("global_load_async_to_lds_b128 %0, %1, off"
               :
               : "v"(lds_off), "v"(gsrc)
               : "memory");
}

__device__ __forceinline__ void wait_async0() {
  asm volatile("s_wait_asynccnt 0x0" ::: "memory");
}

// ---------------------------------------------------------------------------
// Kernel 1: pool the element mask into a [NB, KB] block mask (OR over each
// 32x32 block; matches reference "sum > 0").
// ---------------------------------------------------------------------------
__global__ __launch_bounds__(256) void build_block_mask(
    const int* __restrict__ mask, int* __restrict__ bmask) {
  __shared__ int s[256];
  const int blk = blockIdx.x;          // 0 .. NB*KB-1
  const int nbI = blk >> 7;            // / KB
  const int kbI = blk & (KB - 1);
  const int t = threadIdx.x;

  int acc = 0;
#pragma unroll
  for (int i = 0; i < 4; ++i) {        // 1024 elems / 256 threads
    int idx = i * 256 + t;
    int r = idx >> 5;                  // / 32
    int c = idx & 31;
    acc |= mask[(size_t)(nbI * BH + r) * K_DIM + (kbI * BW + c)];
  }
  s[t] = acc;
  __syncthreads();
#pragma unroll
  for (int off = 128; off > 0; off >>= 1) {
    if (t < off) s[t] |= s[t + off];
    __syncthreads();
  }
  if (t == 0) bmask[blk] = (s[0] != 0) ? 1 : 0;
}

// ---------------------------------------------------------------------------
// Kernel 2: block-sparse GEMM, f32 WMMA, async-LDS double-buffered pipeline.
// ---------------------------------------------------------------------------
__global__ __launch_bounds__(256) void dds_gemm(
    const float* __restrict__ data, const float* __restrict__ weight,
    const int* __restrict__ bmask, float* __restrict__ out) {
  __shared__ float lA[2][TILE_M * LDS_STRIDE];  // 2 x 256 x 36 floats
  __shared__ float lW[2][TILE_N * LDS_STRIDE];  // 2 x  32 x 36 floats

  const int nb    = blockIdx.x;              // n-block index (0..127)
  const int mBase = blockIdx.y * TILE_M;
  const int nBase = nb * TILE_N;
  const int t     = threadIdx.x;
  const int wave  = t >> 5;                  // 0..7
  const int lane  = t & 31;
  const int half  = lane >> 4;               // 0/1 (K-pair / M-row-8 select)
  const int l16   = lane & 15;

  v8f acc[2][2];
#pragma unroll
  for (int i = 0; i < 2; ++i)
#pragma unroll
    for (int j = 0; j < 2; ++j)
      acc[i][j] = (v8f){0.f, 0.f, 0.f, 0.f, 0.f, 0.f, 0.f, 0.f};

  const int* bm = bmask + nb * KB;

  // Per-thread staging coordinates (fixed for all k-blocks).
  const int aRow = t >> 3;                   // 0..31 within each 8-iter step
  const int aC4  = (t & 7) * 4;              // float column of the 16B chunk
  const float* gA = data + (size_t)mBase * K_DIM + aC4;        // + row*K + kBase
  const float* gW = weight + (size_t)(nBase + aRow) * K_DIM + aC4;  // + kBase

  // Issue one k-block's async staging into buffer `b`.
  auto stage = [&](int kb, int b) {
    const int kBase = kb * TILE_K;
#pragma unroll
    for (int it = 0; it < 8; ++it) {
      const int row = it * 32 + aRow;        // 256 rows, 8 per thread
      async_copy_b128(&lA[b][row * LDS_STRIDE + aC4],
                      gA + (size_t)row * K_DIM + kBase);
    }
    async_copy_b128(&lW[b][aRow * LDS_STRIDE + aC4], gW + kBase);
  };

  // Scan to first active k-block (uniform scalar loop).
  int cur = 0;
  while (cur < KB && __builtin_amdgcn_readfirstlane(bm[cur]) == 0) ++cur;
  int buf = 0;
  if (cur < KB) stage(cur, 0);

  const int mW = wave * 32;
  while (cur < KB) {
    // Next active k-block.
    int nxt = cur + 1;
    while (nxt < KB && __builtin_amdgcn_readfirstlane(bm[nxt]) == 0) ++nxt;

    // My async stage of `buf` is done; barrier certifies all waves' stages
    // done AND (from last iteration's compute) that buf^1 is free to refill.
    wait_async0();
    __syncthreads();

    if (nxt < KB) stage(nxt, buf ^ 1);       // overlap fill with compute

    // --- Compute on `buf`: 8 K-steps x 4 WMMAs (2x2 accumulator tiles).
    const float* A = lA[buf];
    const float* W = lW[buf];
#pragma unroll
    for (int kk = 0; kk < TILE_K; kk += 4) {
      const int kc = kk + 2 * half;  // lane-half selects K pair {0,1}/{2,3}
      v2f a0 = *(const v2f*)&A[(mW + l16) * LDS_STRIDE + kc];
      v2f a1 = *(const v2f*)&A[(mW + 16 + l16) * LDS_STRIDE + kc];
      v2f b0 = *(const v2f*)&W[(l16) * LDS_STRIDE + kc];
      v2f b1 = *(const v2f*)&W[(16 + l16) * LDS_STRIDE + kc];

      acc[0][0] = __builtin_amdgcn_wmma_f32_16x16x4_f32(false, a0, false, b0,
                                                        (short)0, acc[0][0],
                                                        false, false);
      acc[0][1] = __builtin_amdgcn_wmma_f32_16x16x4_f32(false, a0, false, b1,
                                                        (short)0, acc[0][1],
                                                        false, false);
      acc[1][0] = __builtin_amdgcn_wmma_f32_16x16x4_f32(false, a1, false, b0,
                                                        (short)0, acc[1][0],
                                                        false, false);
      acc[1][1] = __builtin_amdgcn_wmma_f32_16x16x4_f32(false, a1, false, b1,
                                                        (short)0, acc[1][1],
                                                        false, false);
    }

    cur = nxt;
    buf ^= 1;
  }

  // --- Epilogue: C/D layout -> m = base + v + 8*half, n = base + l16.
#pragma unroll
  for (int i = 0; i < 2; ++i) {
#pragma unroll
    for (int j = 0; j < 2; ++j) {
#pragma unroll
      for (int v = 0; v < 8; ++v) {
        const int m = mBase + wave * 32 + i * 16 + half * 8 + v;
        const int n = nBase + j * 16 + l16;
        out[(size_t)m * N_DIM + n] = acc[i][j][v];
      }
    }
  }
}

// ---------------------------------------------------------------------------
// Host launcher.
//   d_in[0] = data  [4,1024,4096] f32 (flat 4096x4096)
//   d_in[1] = weight[4096,4096]   f32
//   d_in[2] = mask  [4096,4096]   int32
//   d_out   = out   [4,1024,4096] f32
//   d_ws    : >= NB*KB*4 bytes used for the block mask
// ---------------------------------------------------------------------------
extern "C" void kernel_launch(void* const* d_in, const int* in_sizes, int n_in,
                              void* d_out, int out_size, void* d_ws,
                              size_t ws_size, hipStream_t stream) {
  const float* data   = (const float*)d_in[0];
  const float* weight = (const float*)d_in[1];
  const int*   mask   = (const int*)d_in[2];
  float*       out    = (float*)d_out;
  int*         bmask  = (int*)d_ws;

  build_block_mask<<<dim3(NB * KB), 256, 0, stream>>>(mask, bmask);

  dim3 grid(NB, M_DIM / TILE_M);  // (128 n-tiles, 16 m-tiles)
  dds_gemm<<<grid, 256, 0, stream>>>(data, weight, bmask, out);
}